// HeteroGNN_60318520705763
// MI455X (gfx1250) — compile-verified
//
#include <hip/hip_runtime.h>

#define NU   500000
#define NPOI 200000
#define NC   100
#define EUP  500000
#define EPP  1000000
#define HDIM 32

typedef __attribute__((ext_vector_type(2))) float v2f;
typedef __attribute__((ext_vector_type(8))) float v8f;

// ---------------------------------------------------------------------------
// Encoder: out[n,h] = relu(sum_k x[n,k]*W[k,h] + b[h]) (+ emb[n,h] if given)
// ---------------------------------------------------------------------------
__global__ void encode_kernel(const float* __restrict__ x, const float* __restrict__ W,
                              const float* __restrict__ b, const float* __restrict__ emb,
                              float* __restrict__ out, int N, int K) {
    int t = blockIdx.x * blockDim.x + threadIdx.x;
    if (t >= N * HDIM) return;
    int n = t >> 5, h = t & 31;
    float s = b[h];
    for (int k = 0; k < K; ++k) s += x[n * K + k] * W[k * HDIM + h];
    s = fmaxf(s, 0.0f);
    if (emb) s += emb[t];
    out[t] = s;
}

// ---------------------------------------------------------------------------
// Dense transform T[N,32] = A[N,32] @ W[32,32] + b  via V_WMMA_F32_16X16X4_F32
// One wave per 16-row tile; 2 N-halves x 8 K-steps = 16 WMMAs per wave.
// A layout (16x4 f32): lanes 0-15 hold M=0..15 K={0,1}; lanes 16-31 K={2,3}.
// B layout (4x16 f32): mirrored (lane%16 = N-col, lane/16 selects K-half).
// C/D layout (16x16 f32): VGPR v holds M = v + 8*(lane/16), N = lane%16.
// ---------------------------------------------------------------------------
__global__ void __launch_bounds__(128) gemm32_wmma_kernel(
    const float* __restrict__ A, const float* __restrict__ W,
    const float* __restrict__ b, float* __restrict__ T, int N) {
    int wave = threadIdx.x >> 5;
    int lane = threadIdx.x & 31;
    int rowBase = (blockIdx.x * 4 + wave) * 16;
    if (rowBase >= N) return;               // wave-uniform exit: EXEC all-1s below
    int m     = lane & 15;                  // M (for A/C) or N-col (for B/C)
    int khalf = (lane >> 4) << 1;           // 0 or 2
    int row = rowBase + m;
    if (row >= N) row = N - 1;              // clamp loads; stores predicated below
    const float* arow = A + (size_t)row * HDIM;

    for (int nh = 0; nh < 2; ++nh) {
        const float* wcol = W + nh * 16 + m;
        v8f c = {};
#pragma unroll
        for (int kk = 0; kk < 8; ++kk) {
            int kb = kk * 4 + khalf;
            v2f a;  a.x  = arow[kb];            a.y  = arow[kb + 1];
            v2f bb; bb.x = wcol[kb * HDIM];     bb.y = wcol[(kb + 1) * HDIM];
            c = __builtin_amdgcn_wmma_f32_16x16x4_f32(
                    false, a, false, bb, (short)0, c, false, false);
        }
        int mbase = (lane >> 4) * 8;
        float bias = b[nh * 16 + m];
#pragma unroll
        for (int v = 0; v < 8; ++v) {
            int r = rowBase + mbase + v;
            if (r < N) T[(size_t)r * HDIM + nh * 16 + m] = c[v] + bias;
        }
    }
}

// ---------------------------------------------------------------------------
// Zero accumulators
// ---------------------------------------------------------------------------
__global__ void zero_kernel(float* __restrict__ acc, float* __restrict__ cnt, int Ndst) {
    int t = blockIdx.x * blockDim.x + threadIdx.x;
    if (t < Ndst * HDIM) acc[t] = 0.0f;
    if (t < Ndst) cnt[t] = 0.0f;
}

// ---------------------------------------------------------------------------
// Edge scatter: acc[dst] += T[src] * ea[e];  cnt[dst] += 1
// 8 threads per edge, float4 per thread -> global_atomic_add_f32
// ---------------------------------------------------------------------------
__global__ void scatter_kernel(const float* __restrict__ T, const int* __restrict__ ei,
                               const float* __restrict__ ea, float* __restrict__ acc,
                               float* __restrict__ cnt, int E) {
    int t = blockIdx.x * blockDim.x + threadIdx.x;
    if (t >= E * 8) return;
    int e = t >> 3, j = t & 7;
    int src = ei[e];
    int dst = ei[E + e];
    float4 v = *(const float4*)(T + (size_t)src * HDIM + j * 4);
    float w = ea ? ea[e] : 1.0f;
    float* a = acc + (size_t)dst * HDIM + j * 4;
    atomicAdd(a + 0, v.x * w);
    atomicAdd(a + 1, v.y * w);
    atomicAdd(a + 2, v.z * w);
    atomicAdd(a + 3, v.w * w);
    if (j == 0) atomicAdd(cnt + dst, 1.0f);
}

// ---------------------------------------------------------------------------
// Finalize: dst += relu(acc / max(cnt,1))
// ---------------------------------------------------------------------------
__global__ void finalize_kernel(const float* __restrict__ acc, const float* __restrict__ cnt,
                                float* __restrict__ dst, int Ndst) {
    int t = blockIdx.x * blockDim.x + threadIdx.x;
    if (t >= Ndst * HDIM) return;
    int n = t >> 5;
    float c = fmaxf(cnt[n], 1.0f);
    dst[t] += fmaxf(acc[t] / c, 0.0f);
}

// ---------------------------------------------------------------------------

static inline void launch_gemm(const float* src, const float* W, const float* b,
                               float* T, int N, hipStream_t s) {
    gemm32_wmma_kernel<<<(N + 63) / 64, 128, 0, s>>>(src, W, b, T, N);
}

static inline void launch_prop(const float* T, const int* ei, const float* ea, int E,
                               float* dstH, int Ndst, float* acc, float* cnt,
                               hipStream_t s) {
    int zt = Ndst * HDIM;
    zero_kernel<<<(zt + 255) / 256, 256, 0, s>>>(acc, cnt, Ndst);
    int st = E * 8;
    scatter_kernel<<<(st + 255) / 256, 256, 0, s>>>(T, ei, ea, acc, cnt, E);
    finalize_kernel<<<(zt + 255) / 256, 256, 0, s>>>(acc, cnt, dstH, Ndst);
}

extern "C" void kernel_launch(void* const* d_in, const int* in_sizes, int n_in,
                              void* d_out, int out_size, void* d_ws, size_t ws_size,
                              hipStream_t stream) {
    (void)in_sizes; (void)n_in; (void)out_size; (void)ws_size;

    // ---- inputs (setup_inputs dict order) ----
    const float* x_user = (const float*)d_in[0];
    const float* x_poi  = (const float*)d_in[1];
    const float* x_cate = (const float*)d_in[2];
    // 3..24: [user_lin, poi_lin, cate_lin, W1_u, W1_p, W1_c, W1_pp,
    //         W2_u, W2_p, W2_c, W2_pp] each as (_w, _b)
    const float* user_lin_w = (const float*)d_in[3];
    const float* user_lin_b = (const float*)d_in[4];
    const float* poi_lin_w  = (const float*)d_in[5];
    const float* poi_lin_b  = (const float*)d_in[6];
    const float* cate_lin_w = (const float*)d_in[7];
    const float* cate_lin_b = (const float*)d_in[8];
    const float* cate_emb   = (const float*)d_in[25];
    const int* ei_pv        = (const int*)d_in[26];
    const int* ei_mc        = (const int*)d_in[27];
    const int* ei_order     = (const int*)d_in[28];
    const int* ei_rev_pv    = (const int*)d_in[29];
    const int* ei_rev_mc    = (const int*)d_in[30];
    const int* ei_rev_order = (const int*)d_in[31];
    const int* ei_belongs   = (const int*)d_in[32];
    const int* ei_pp        = (const int*)d_in[33];
    const float* ea_pv        = (const float*)d_in[34];
    const float* ea_mc        = (const float*)d_in[35];
    const float* ea_order     = (const float*)d_in[36];
    const float* ea_rev_pv    = (const float*)d_in[37];
    const float* ea_rev_mc    = (const float*)d_in[38];
    const float* ea_rev_order = (const float*)d_in[39];
    const float* ea_pp        = (const float*)d_in[40];

    // ---- output layout: [user_h | poi_h | cate_h], computed in place ----
    float* user_h = (float*)d_out;
    float* poi_h  = user_h + (size_t)NU * HDIM;
    float* cate_h = poi_h + (size_t)NPOI * HDIM;

    // ---- workspace: T (NU*32) | acc (NU*32) | cnt (NU) ----
    float* T   = (float*)d_ws;
    float* acc = T + (size_t)NU * HDIM;
    float* cnt = acc + (size_t)NU * HDIM;

    // ---- encoders ----
    encode_kernel<<<(NU * HDIM + 255) / 256, 256, 0, stream>>>(
        x_user, user_lin_w, user_lin_b, nullptr, user_h, NU, 2);
    encode_kernel<<<(NPOI * HDIM + 255) / 256, 256, 0, stream>>>(
        x_poi, poi_lin_w, poi_lin_b, nullptr, poi_h, NPOI, 3);
    encode_kernel<<<(NC * HDIM + 255) / 256, 256, 0, stream>>>(
        x_cate, cate_lin_w, cate_lin_b, cate_emb, cate_h, NC, 1);

    // ---- 2 message-passing layers ----
    for (int L = 0; L < 2; ++L) {
        int base = 9 + L * 8;
        const float* Wu  = (const float*)d_in[base + 0];
        const float* bu  = (const float*)d_in[base + 1];
        const float* Wp  = (const float*)d_in[base + 2];
        const float* bp  = (const float*)d_in[base + 3];
        const float* Wc  = (const float*)d_in[base + 4];
        const float* bc  = (const float*)d_in[base + 5];
        const float* Wpp = (const float*)d_in[base + 6];
        const float* bpp = (const float*)d_in[base + 7];

        // poi -> user (3 props share the same transform T = poi_h @ Wp + bp)
        launch_gemm(poi_h, Wp, bp, T, NPOI, stream);
        launch_prop(T, ei_rev_pv,    ea_rev_pv,    EUP, user_h, NU, acc, cnt, stream);
        launch_prop(T, ei_rev_mc,    ea_rev_mc,    EUP, user_h, NU, acc, cnt, stream);
        launch_prop(T, ei_rev_order, ea_rev_order, EUP, user_h, NU, acc, cnt, stream);

        // user -> poi (3 props share T = user_h @ Wu + bu)
        launch_gemm(user_h, Wu, bu, T, NU, stream);
        launch_prop(T, ei_pv,    ea_pv,    EUP, poi_h, NPOI, acc, cnt, stream);
        launch_prop(T, ei_mc,    ea_mc,    EUP, poi_h, NPOI, acc, cnt, stream);
        launch_prop(T, ei_order, ea_order, EUP, poi_h, NPOI, acc, cnt, stream);

        // cate -> poi (no edge weights)
        launch_gemm(cate_h, Wc, bc, T, NC, stream);
        launch_prop(T, ei_belongs, nullptr, NPOI, poi_h, NPOI, acc, cnt, stream);

        // poi -> poi (source is poi_h AFTER the updates above, as in reference)
        launch_gemm(poi_h, Wpp, bpp, T, NPOI, stream);
        launch_prop(T, ei_pp, ea_pp, EPP, poi_h, NPOI, acc, cnt, stream);
    }
}